// DriveAttention_71365176590623
// MI455X (gfx1250) — compile-verified
//
#include <hip/hip_runtime.h>
#include <hip/hip_bf16.h>
#include <math.h>

typedef float v2f __attribute__((ext_vector_type(2)));
typedef float v4f __attribute__((ext_vector_type(4)));
typedef float v8f __attribute__((ext_vector_type(8)));

#define BS_ 16
#define N_  2048
#define T_  3

// ---------------------------------------------------------------------------
// GEMM: Y(16,Nout) = act( X(16,K) @ W(Nout,K)^T + bias [+ R] )
// One block per 16-column tile. NW waves split K; each wave accumulates a
// 16x16 f32 C tile with V_WMMA_F32_16X16X4_F32; partials reduced via LDS.
// Each lane loads b128 (4 K-values for its half); two WMMAs consume the 8 K
// under the permutation k = kk + 4*half + 2*step + j (identical for A and B,
// so the K-reduction is exact).
// ---------------------------------------------------------------------------
template<int NW>
__global__ __launch_bounds__(NW * 32) void gemm16_wmma(
    const float* __restrict__ X, int sx,
    const float* __restrict__ W,
    const float* __restrict__ bias,
    float* __restrict__ Y, int sy,
    const float* __restrict__ R, int sr,
    int K, int relu)
{
    __shared__ float red[NW * 256];
    const int tile = blockIdx.x;
    const int tid  = threadIdx.x;
    const int lane = tid & 31;
    const int wv   = tid >> 5;
    const int kchunk = K / NW;          // always a multiple of 8 here
    const int k0   = wv * kchunk;
    const int quad = lane >> 4;         // half: 0 -> K base +0, 1 -> +4
    const int r16  = lane & 15;         // A: row M ; B: col N

    const float* Ap = X + (size_t)r16 * sx + k0 + 4 * quad;
    const float* Bp = W + (size_t)(tile * 16 + r16) * K + k0 + 4 * quad;

    v8f c = {0.f,0.f,0.f,0.f,0.f,0.f,0.f,0.f};
    #pragma unroll 4
    for (int kk = 0; kk < kchunk; kk += 8) {
        v4f a = *(const v4f*)(Ap + kk);
        v4f b = *(const v4f*)(Bp + kk);
        v2f a0 = {a.x, a.y}, b0 = {b.x, b.y};
        v2f a1 = {a.z, a.w}, b1 = {b.z, b.w};
        c = __builtin_amdgcn_wmma_f32_16x16x4_f32(
                false, a0, false, b0, (short)0, c, false, false);
        c = __builtin_amdgcn_wmma_f32_16x16x4_f32(
                false, a1, false, b1, (short)0, c, false, false);
    }

    #pragma unroll
    for (int i = 0; i < 8; ++i) red[wv * 256 + lane * 8 + i] = c[i];
    __syncthreads();

    if (tid < 256) {
        float sum = 0.f;
        #pragma unroll
        for (int w = 0; w < NW; ++w) sum += red[w * 256 + tid];

        // tid -> (lane l, vgpr i) -> (m, n): C layout M = i + 8*(l/16), N = l%16
        const int l = tid >> 3, i = tid & 7;
        const int m = i + 8 * (l >> 4);
        const int n = tile * 16 + (l & 15);
        float val = sum + bias[n];
        if (R) val += R[(size_t)m * sr + n];
        if (relu) val = fmaxf(val, 0.f);
        Y[(size_t)m * sy + n] = val;
    }
}

// ---------------------------------------------------------------------------
// Rank-1 attention (f==1):
//   out[b,n] = sum_m softmax_m( s_n * k[b,m] ) * v[b,m]  (+ residual)
// v row is staged into LDS with CDNA5 async copies (ASYNCcnt) while k is
// loaded normally and reduced for max/min. Diagonal mask handled by
// subtracting the m==n term (unmasked analytic max upper-bounds masked max).
// ---------------------------------------------------------------------------
__global__ __launch_bounds__(256) void attn_f1(
    const float* __restrict__ q,
    const float* __restrict__ k,
    const float* __restrict__ v,
    const float* __restrict__ R, int sr,
    float* __restrict__ Y, int sy,
    int maskdiag, float scale)
{
    __shared__ float ks[N_];
    __shared__ float vs[N_];
    __shared__ float rmax[256];
    __shared__ float rmin[256];
    const int b   = blockIdx.y;
    const int tid = threadIdx.x;
    const float* kb = k + (size_t)b * N_;
    const float* vb = v + (size_t)b * N_;

    // async-copy v[b,:] (8KB) into LDS: global_load_async_to_lds_b128
    {
        unsigned vbase = (unsigned)(uintptr_t)(&vs[0]);
        #pragma unroll
        for (int it = 0; it < 2; ++it) {
            int m4 = (tid + it * 256) * 4;
            unsigned ldsa = vbase + (unsigned)m4 * 4u;
            const float* ga = vb + m4;
            asm volatile("global_load_async_to_lds_b128 %0, %1, off"
                         :: "v"(ldsa), "v"(ga) : "memory");
        }
    }

    float lmax = -3.0e38f, lmin = 3.0e38f;
    for (int m = tid; m < N_; m += 256) {
        float kk = kb[m];
        ks[m] = kk;
        lmax = fmaxf(lmax, kk);
        lmin = fminf(lmin, kk);
    }
    rmax[tid] = lmax; rmin[tid] = lmin;
    asm volatile("s_wait_asynccnt 0x0" ::: "memory");
    __syncthreads();
    for (int s2 = 128; s2 > 0; s2 >>= 1) {
        if (tid < s2) {
            rmax[tid] = fmaxf(rmax[tid], rmax[tid + s2]);
            rmin[tid] = fminf(rmin[tid], rmin[tid + s2]);
        }
        __syncthreads();
    }
    const float kmax = rmax[0], kmin = rmin[0];

    const int   n  = blockIdx.x * 256 + tid;
    const float s  = q[(size_t)b * N_ + n] * scale;
    const float mx = (s >= 0.f) ? s * kmax : s * kmin;

    float sum = 0.f, acc = 0.f;
    #pragma unroll 4
    for (int m = 0; m < N_; ++m) {
        float w = __expf(fmaf(s, ks[m], -mx));
        sum += w;
        acc = fmaf(w, vs[m], acc);
    }
    if (maskdiag) {
        float wd = __expf(fmaf(s, ks[n], -mx));
        sum -= wd;
        acc = fmaf(-wd, vs[n], acc);
    }
    float val = acc / sum;
    if (R) val += R[(size_t)b * sr + n];
    Y[(size_t)b * sy + n] = val;
}

// m_hiss_out[b,t,d] = relu( m2cat[b, d*3 + t] )
__global__ void hiss_out_k(const float* __restrict__ cat, float* __restrict__ out)
{
    int idx = blockIdx.x * 256 + threadIdx.x;
    if (idx >= BS_ * T_ * N_) return;
    int b = idx / (T_ * N_);
    int r = idx - b * (T_ * N_);
    int t = r / N_;
    int d = r - t * N_;
    out[idx] = fmaxf(cat[(size_t)b * (T_ * N_) + d * T_ + t], 0.f);
}

// ---------------------------------------------------------------------------
struct PP { const float* w; const float* b; };

extern "C" void kernel_launch(void* const* d_in, const int* in_sizes, int n_in,
                              void* d_out, int out_size, void* d_ws, size_t ws_size,
                              hipStream_t stream)
{
    (void)out_size; (void)ws_size;
    const float* m0    = (const float*)d_in[0];
    const float* mhiss = (const float*)d_in[1];

    PP q0,k0,v0,q1,k_his,v_his,f11t,f12t,f21t,f22t,f11st,f12st,f21st,f22st,
       q2,k2,v2,m11,m12,mh1,mh2,m31,m32;

    if (n_in >= 48 && in_sizes[2] == 2048 * 2048) {
        // insertion order, w then b
        int i = 2;
        auto take = [&](PP& p){ p.w = (const float*)d_in[i]; p.b = (const float*)d_in[i+1]; i += 2; };
        take(q0); take(k0); take(v0); take(q1); take(k_his); take(v_his);
        take(f11t); take(f12t); take(f21t); take(f22t);
        take(f11st); take(f12st); take(f21st); take(f22st);
        take(q2); take(k2); take(v2);
        take(m11); take(m12); take(mh1); take(mh2); take(m31); take(m32);
    } else if (n_in >= 48) {
        // jax tree_flatten: keys sorted, 'b' before 'w'
        int i = 2;
        auto take = [&](PP& p){ p.b = (const float*)d_in[i]; p.w = (const float*)d_in[i+1]; i += 2; };
        take(f11st); take(f11t); take(f12st); take(f12t);
        take(f21st); take(f21t); take(f22st); take(f22t);
        take(k0); take(k2); take(k_his);
        take(m11); take(m12); take(m31); take(m32); take(mh1); take(mh2);
        take(q0); take(q1); take(q2);
        take(v0); take(v2); take(v_his);
    } else {
        // single packed blob, insertion order (w then b)
        const float* p = (const float*)d_in[2];
        auto take = [&](PP& pp, int fin, int fout){ pp.w = p; p += (size_t)fin*fout; pp.b = p; p += fout; };
        take(q0,2048,2048); take(k0,2048,2048); take(v0,2048,2048); take(q1,2048,2048);
        take(k_his,2048,2048); take(v_his,2048,2048);
        take(f11t,6144,2048); take(f12t,2048,2048); take(f21t,2048,2048); take(f22t,2048,2048);
        take(f11st,4096,2048); take(f12st,2048,2048); take(f21st,2048,2048); take(f22st,2048,2048);
        take(q2,2048,2048); take(k2,2048,2048); take(v2,2048,2048);
        take(m11,2048,512); take(m12,512,2048); take(mh1,2048,512); take(mh2,512,2048);
        take(m31,2048,512); take(m32,512,2048);
    }

    float* ws = (float*)d_ws;
    float* bq    = ws; ws += 16 * 2048;
    float* bk    = ws; ws += 16 * 2048;
    float* bv    = ws; ws += 16 * 2048;
    float* bh    = ws; ws += 16 * 512;
    float* ba    = ws; ws += 16 * 2048;
    float* bcat  = ws; ws += 16 * 6144;   // m2cat (row stride 6144)
    float* bfuse = ws; ws += 16 * 4096;   // [skip1 | skip2] (row stride 4096)
    float* bmf   = ws; ws += 16 * 2048;
    float* bm3   = ws; ws += 16 * 2048;

    float* outm3 = (float*)d_out;
    float* outh  = (float*)d_out + 16 * 2048;

    auto gemm = [&](const float* X, int sx, const PP& P, float* Y, int sy,
                    const float* R, int sr, int K, int Nout, int relu) {
        if (Nout >= 2048)
            gemm16_wmma<8><<<Nout / 16, 256, 0, stream>>>(X, sx, P.w, P.b, Y, sy, R, sr, K, relu);
        else
            gemm16_wmma<32><<<Nout / 16, 1024, 0, stream>>>(X, sx, P.w, P.b, Y, sy, R, sr, K, relu);
    };
    auto attn = [&](const float* q_, const float* k_, const float* v_,
                    const float* R, int sr, float* Y, int sy, int mask) {
        attn_f1<<<dim3(8, 16), 256, 0, stream>>>(q_, k_, v_, R, sr, Y, sy, mask, 1.0f);
    };

    // ---- block 0: self attention on m_0, masked diagonal ----
    gemm(m0, 2048, q0, bq, 2048, nullptr, 0, 2048, 2048, 0);
    gemm(m0, 2048, k0, bk, 2048, nullptr, 0, 2048, 2048, 0);
    gemm(m0, 2048, v0, bv, 2048, nullptr, 0, 2048, 2048, 0);
    attn(bq, bk, bv, m0, 2048, bfuse, 4096, 1);                     // m1_pre -> skip1 slot
    gemm(bfuse, 4096, m11, bh, 512, nullptr, 0, 2048, 512, 1);
    gemm(bh, 512, m12, bfuse, 4096, bfuse, 4096, 512, 2048, 0);     // m1 (= skip1)

    gemm(bfuse, 4096, q1, bq, 2048, nullptr, 0, 2048, 2048, 0);     // q1

    // ---- history attention, t = 0..2 ----
    for (int t = 0; t < 3; ++t) {
        const float* mh = mhiss + t * 2048;                          // row stride 6144
        float* m2 = bcat + t * 2048;                                 // column slice of m2cat
        gemm(mh, 6144, k_his, bk, 2048, nullptr, 0, 2048, 2048, 0);
        gemm(mh, 6144, v_his, bv, 2048, nullptr, 0, 2048, 2048, 0);
        attn(bq, bk, bv, mh, 6144, m2, 6144, 0);
        gemm(m2, 6144, mh1, bh, 512, nullptr, 0, 2048, 512, 1);
        gemm(bh, 512, mh2, m2, 6144, m2, 6144, 512, 2048, 0);
    }
    hiss_out_k<<<(BS_ * T_ * N_) / 256, 256, 0, stream>>>(bcat, outh);

    // ---- temporal fuse ----
    gemm(bcat, 6144, f11t, ba, 2048, nullptr, 0, 6144, 2048, 1);
    gemm(ba, 2048, f12t, bfuse + 2048, 4096, nullptr, 0, 2048, 2048, 0);
    gemm(bfuse + 2048, 4096, f21t, ba, 2048, nullptr, 0, 2048, 2048, 1);
    gemm(ba, 2048, f22t, bfuse + 2048, 4096, bfuse + 2048, 4096, 2048, 2048, 0); // skip2

    // ---- skip fuse (input = [skip1|skip2], contiguous rows of 4096) ----
    gemm(bfuse, 4096, f11st, ba, 2048, nullptr, 0, 4096, 2048, 1);
    gemm(ba, 2048, f12st, bmf, 2048, nullptr, 0, 2048, 2048, 0);
    gemm(bmf, 2048, f21st, ba, 2048, nullptr, 0, 2048, 2048, 1);
    gemm(ba, 2048, f22st, bmf, 2048, bmf, 2048, 2048, 2048, 0);      // m_fuse

    // ---- final attention + MLP + relu ----
    gemm(bmf, 2048, q2, bq, 2048, nullptr, 0, 2048, 2048, 0);
    gemm(bmf, 2048, k2, bk, 2048, nullptr, 0, 2048, 2048, 0);
    gemm(bmf, 2048, v2, bv, 2048, nullptr, 0, 2048, 2048, 0);
    attn(bq, bk, bv, bmf, 2048, bm3, 2048, 0);
    gemm(bm3, 2048, m31, bh, 512, nullptr, 0, 2048, 512, 1);
    gemm(bh, 512, m32, outm3, 2048, bm3, 2048, 512, 2048, 1);        // m3 -> d_out
}